// NKNN_48722109006281
// MI455X (gfx1250) — compile-verified
//
#include <hip/hip_runtime.h>

#define BN 8
#define DN 256
#define NN 32768
#define FN 256
#define KN 16
#define NSEG 4
#define SEGN (NN / NSEG)     // 8192 n per aggregation block
#define CHK 128              // n chunk per double-buffer stage
#define VSTR 132             // padded LDS row stride for values tile

typedef __attribute__((ext_vector_type(2))) float v2f;
typedef __attribute__((ext_vector_type(8))) float v8f;
typedef int v4i __attribute__((__vector_size__(4 * sizeof(int))));

#define GLOBAL_AS __attribute__((address_space(1)))
#define LDS_AS __attribute__((address_space(3)))

#if __has_builtin(__builtin_amdgcn_global_load_async_to_lds_b128)
#define HAVE_ASYNC_LDS 1
#else
#define HAVE_ASYNC_LDS 0
#endif

#if __has_builtin(__builtin_amdgcn_s_wait_asynccnt)
#define WAIT_ASYNC(n) __builtin_amdgcn_s_wait_asynccnt(n)
#else
#define WAIT_ASYNC(n) asm volatile("s_wait_asynccnt %0" ::"i"(n) : "memory")
#endif

// ---------------- Kernel 1: sims[b,n] = -(q[b,:] . keys[b,:,n]) / 16 ----------------
__global__ void nknn_sims_kernel(const float* __restrict__ q,
                                 const float* __restrict__ keys,
                                 float* __restrict__ sims) {
  __shared__ float qs[DN];
  const int b = blockIdx.y;
  const int n = blockIdx.x * 256 + threadIdx.x;
  qs[threadIdx.x] = q[b * DN + threadIdx.x];   // blockDim == DN == 256
  __syncthreads();
  const float* kp = keys + (size_t)b * DN * NN + n;
  float acc = 0.f;
#pragma unroll 8
  for (int d = 0; d < DN; ++d)
    acc += qs[d] * kp[(size_t)d * NN];
  sims[b * NN + n] = -acc * (1.0f / 16.0f);
}

// ---------------- wave32 / 16-wave block reductions (blockDim == 512) ----------------
__device__ __forceinline__ float warpMax(float v) {
#pragma unroll
  for (int o = 16; o > 0; o >>= 1) v = fmaxf(v, __shfl_xor(v, o, 32));
  return v;
}
__device__ __forceinline__ float warpSum(float v) {
#pragma unroll
  for (int o = 16; o > 0; o >>= 1) v += __shfl_xor(v, o, 32);
  return v;
}
__device__ __forceinline__ float blockMax16(float v, float* red) {
  const int lane = threadIdx.x & 31, wid = threadIdx.x >> 5;
  v = warpMax(v);
  if (lane == 0) red[wid] = v;
  __syncthreads();
  if (wid == 0) {
    float x = (lane < 16) ? red[lane] : -1e30f;
    x = warpMax(x);
    if (lane == 0) red[0] = x;
  }
  __syncthreads();
  float r = red[0];
  __syncthreads();
  return r;
}
__device__ __forceinline__ float blockSum16(float v, float* red) {
  const int lane = threadIdx.x & 31, wid = threadIdx.x >> 5;
  v = warpSum(v);
  if (lane == 0) red[wid] = v;
  __syncthreads();
  if (wid == 0) {
    float x = (lane < 16) ? red[lane] : 0.f;
    x = warpSum(x);
    if (lane == 0) red[0] = x;
  }
  __syncthreads();
  float r = red[0];
  __syncthreads();
  return r;
}

// ---------------- Kernel 2: iterative softmax, omega[b,n,k] ----------------
// TEMP==1 closed form: G = exp(alpha); omega_k = G/sum(G); alpha += log(1-omega)
// is exactly G *= (1-omega). No exp/log in the K loop; G register-resident.
__global__ void __launch_bounds__(512) nknn_omega_kernel(
    const float* __restrict__ sims, float* __restrict__ omega) {
  __shared__ float red[16];
  const int b = blockIdx.x;
  const int t = threadIdx.x;
  const float* sb = sims + (size_t)b * NN;
  float* ob = omega + (size_t)b * NN * KN;

  float G[64];
#pragma unroll
  for (int j = 0; j < 64; ++j) G[j] = sb[t + j * 512];

  float m = -1e30f;
#pragma unroll
  for (int j = 0; j < 64; ++j) m = fmaxf(m, G[j]);
  m = blockMax16(m, red);
  float s = 0.f;
#pragma unroll
  for (int j = 0; j < 64; ++j) { G[j] = __expf(G[j] - m); s += G[j]; }
  s = blockSum16(s, red);
  const float inv0 = 1.0f / s;
#pragma unroll
  for (int j = 0; j < 64; ++j) G[j] = __expf(G[j] * inv0);  // G = exp(alpha0)

  for (int k = 0; k < KN; ++k) {
    float sk = 0.f;
#pragma unroll
    for (int j = 0; j < 64; ++j) sk += G[j];
    sk = blockSum16(sk, red);
    const float inv = 1.0f / sk;
#pragma unroll
    for (int j = 0; j < 64; ++j) {
      const float om = G[j] * inv;
      ob[(size_t)(t + j * 512) * KN + k] = om;
      G[j] *= (1.0f - om);
    }
  }
}

// ---------------- staging helpers (async global->LDS, with sync fallback) ----------------
__device__ __forceinline__ void copy_b128(const float* g, float* l) {
#if HAVE_ASYNC_LDS
  __builtin_amdgcn_global_load_async_to_lds_b128((GLOBAL_AS v4i*)g,
                                                 (LDS_AS v4i*)l, 0, 0);
#else
  float4 x = *(const float4*)g;
  *(float4*)l = x;
#endif
}

// ---------------- Kernel 3: partial[seg,b,k,f] = sum_{n in seg} omega[b,n,k]*values[b,f,n] ----
// Per-batch GEMM (16 x N) x (N x 256) via V_WMMA_F32_16X16X4_F32 with
// double-buffered GLOBAL_LOAD_ASYNC_TO_LDS_B128 pipelining (ASYNCcnt-tracked).
// Grid (16 F-tiles, 8 batches, NSEG n-segments); block = 256 threads = 8 waves.
__global__ void __launch_bounds__(256) nknn_agg_kernel(
    const float* __restrict__ omega, const float* __restrict__ values,
    float* __restrict__ partial) {
  __shared__ __align__(16) float lds_v[2][16 * VSTR];  // values tile [16 f x CHK n]
  __shared__ __align__(16) float lds_o[2][CHK * 16];   // omega tile  [CHK n x 16 k]
  __shared__ __align__(16) float lds_red[8 * 256];     // per-wave partial C tiles

  const int ft = blockIdx.x;           // F tile 0..15
  const int b = blockIdx.y;            // batch
  const int seg = blockIdx.z;          // n segment
  const int f_base = ft * 16;
  const int nbase = seg * SEGN;
  const int t = threadIdx.x;           // 0..255
  const int lane = t & 31;
  const int w = t >> 5;                // wave 0..7
  const int mcol = lane & 15;          // A row (k idx) == B col (f idx)
  const int j0 = (lane >> 4) * 2;      // Kc sub-column pair per lane half

  const float* vb = values + ((size_t)b * FN + f_base) * NN + nbase;
  const float* obase = omega + ((size_t)b * NN + nbase) * KN;

  v8f c = {};                          // f32 16x16 accumulator (8 VGPRs)

  const int r = t >> 4;                // values staging row 0..15
  const int c0 = (t & 15) * 8;         // values staging col start (8 floats/thread)

  // issue one chunk's copies: 2 b128 for values + 2 b128 for omega per thread
  auto issue = [&](int buf, int nc) {
    const float* src = vb + (size_t)r * NN + nc + c0;
    float* dv = &lds_v[buf][r * VSTR + c0];
    copy_b128(src + 0, dv + 0);
    copy_b128(src + 4, dv + 4);
    const float* osrc = obase + (size_t)nc * KN + t * 8;
    float* dst_o = &lds_o[buf][t * 8];
    copy_b128(osrc + 0, dst_o + 0);
    copy_b128(osrc + 4, dst_o + 4);
#if !HAVE_ASYNC_LDS
    __builtin_amdgcn_sched_barrier(0);
#endif
  };

  const int nchunks = SEGN / CHK;      // 64
  issue(0, 0);

  for (int i = 0; i < nchunks; ++i) {
    const int cur = i & 1;
    if (i + 1 < nchunks) {
      issue(1 - cur, (i + 1) * CHK);
      WAIT_ASYNC(4);                   // async ops complete in order: chunk i done
    } else {
      WAIT_ASYNC(0);
    }
    __syncthreads();

    const float* lv = lds_v[cur];
    const float* lo = lds_o[cur];
    const int nl_base = w * (CHK / 8);
#pragma unroll
    for (int s = 0; s < CHK / 32; ++s) {
      const int nl = nl_base + s * 4;
      // A (16x4, M=k idx, Kc=n): A[m,j] = lo[(nl+j)*16 + m]
      const float* ap = &lo[(nl + j0) * 16 + mcol];
      v2f a;
      a.x = ap[0];
      a.y = ap[16];
      // B (4x16, Kc=n, N=f): B[j,f] = lv[f*VSTR + nl + j]
      const float* bp = &lv[mcol * VSTR + nl + j0];
      v2f bb;
      bb.x = bp[0];
      bb.y = bp[1];
      c = __builtin_amdgcn_wmma_f32_16x16x4_f32(false, a, false, bb,
                                                (short)0, c, false, false);
    }
    __syncthreads();  // all waves done with buf cur before it is refilled
  }

  // reduce 8 per-wave C tiles via LDS, write partial[seg, b, k, f_base+f]
#pragma unroll
  for (int rr = 0; rr < 8; ++rr) {
    const int mm = rr + 8 * (lane >> 4);
    const int nn2 = lane & 15;
    lds_red[w * 256 + mm * 16 + nn2] = c[rr];
  }
  __syncthreads();
  float acc = 0.f;
#pragma unroll
  for (int ww = 0; ww < 8; ++ww) acc += lds_red[ww * 256 + t];
  const int mo = t >> 4;   // k neighbor index
  const int no = t & 15;   // f within tile
  partial[(((size_t)seg * BN + b) * KN + mo) * FN + f_base + no] = acc;
}

// ---------------- Kernel 4: deterministic reduction of NSEG partials ----------------
__global__ void nknn_reduce_kernel(const float* __restrict__ partial,
                                   float* __restrict__ out) {
  const int i = blockIdx.x * 256 + threadIdx.x;   // 0 .. B*K*F-1
  float acc = 0.f;
#pragma unroll
  for (int s = 0; s < NSEG; ++s)
    acc += partial[(size_t)s * (BN * KN * FN) + i];
  out[i] = acc;
}

extern "C" void kernel_launch(void* const* d_in, const int* in_sizes, int n_in,
                              void* d_out, int out_size, void* d_ws, size_t ws_size,
                              hipStream_t stream) {
  (void)in_sizes; (void)n_in; (void)out_size; (void)ws_size;
  const float* q = (const float*)d_in[0];
  const float* keys = (const float*)d_in[1];
  const float* values = (const float*)d_in[2];
  float* out = (float*)d_out;

  float* sims = (float*)d_ws;                              // B*N floats   (1 MB)
  float* omega = sims + (size_t)BN * NN;                   // B*N*K floats (16 MB)
  float* partial = omega + (size_t)BN * NN * KN;           // NSEG*B*K*F   (0.5 MB)

  nknn_sims_kernel<<<dim3(NN / 256, BN), 256, 0, stream>>>(q, keys, sims);
  nknn_omega_kernel<<<dim3(BN), 512, 0, stream>>>(sims, omega);
  nknn_agg_kernel<<<dim3(FN / 16, BN, NSEG), 256, 0, stream>>>(omega, values, partial);
  nknn_reduce_kernel<<<dim3((BN * KN * FN) / 256), 256, 0, stream>>>(partial, out);
}